// ConvolutionalReverb_5368709120010
// MI455X (gfx1250) — compile-verified
//
#include <hip/hip_runtime.h>
#include <math.h>

// ---------------------------------------------------------------------------
// Convolutional reverb = FFT convolution (same math as the reference):
//   nfft = 2^20, four-step FFT (1024 x 1024), Stockham radix-4 in LDS,
//   pointwise  Y[b,o,f] = sum_ci X[b,ci,f] * conj(W[o,ci,f])  done as a
//   V_WMMA_F32_16X16X4_F32 per frequency bin (M=batch=16, K=2ci*(re,im)=4).
// Workspace: 36 signals * 2^20 complex f32 = 288 MiB in d_ws (einsum in-place).
// Global<->LDS tile staging uses GLOBAL_*_ASYNC_*_LDS_B64 when the toolchain
// declares the builtins (ASYNCcnt path), else plain loads/stores.
// ---------------------------------------------------------------------------

#define NFFT      (1 << 20)
#define NFFT_MASK (NFFT - 1)
#define N1        1024
#define N2        1024
#define BATCH     16
#define CI        2
#define CO        2
#define T_LEN     480000
#define K_LEN     143999
#define NSIG_X    (BATCH * CI)      // 32 input signals
#define NSIG_W    (CO * CI)         // 4  IR signals
#define NSIG      (NSIG_X + NSIG_W) // 36
#define OUT_LEN   (T_LEN + 1)       // 480001

typedef float v2f __attribute__((ext_vector_type(2)));
typedef float v8f __attribute__((ext_vector_type(8)));

// ----- optional async global<->LDS copy path (gfx1250, ASYNCcnt) -----------
#ifdef __has_builtin
#if __has_builtin(__builtin_amdgcn_global_load_async_to_lds_b64) &&   \
    __has_builtin(__builtin_amdgcn_global_store_async_from_lds_b64) && \
    __has_builtin(__builtin_amdgcn_s_wait_asynccnt)
#define HAVE_ASYNC_LDS 1
#endif
#endif
#ifndef HAVE_ASYNC_LDS
#define HAVE_ASYNC_LDS 0
#endif

#if HAVE_ASYNC_LDS
// b64 payload type per clang's builtin signature: int __vector_size__(8)
typedef int a64v __attribute__((vector_size(8)));
typedef __attribute__((address_space(3))) a64v* lds_a64p;
#define ASYNC_LOAD_B64(gsrc, ldst)                                         \
  __builtin_amdgcn_global_load_async_to_lds_b64((a64v*)(gsrc),             \
                                                (lds_a64p)(ldst), 0, 0)
#define ASYNC_STORE_B64(gdst, lsrc)                                        \
  __builtin_amdgcn_global_store_async_from_lds_b64((a64v*)(gdst),          \
                                                   (lds_a64p)(lsrc), 0, 0)
#endif

__device__ __forceinline__ float2 cadd(float2 a, float2 b) {
  return make_float2(a.x + b.x, a.y + b.y);
}
__device__ __forceinline__ float2 csub(float2 a, float2 b) {
  return make_float2(a.x - b.x, a.y - b.y);
}
__device__ __forceinline__ float2 cmul(float2 a, float2 b) {
  return make_float2(fmaf(a.x, b.x, -a.y * b.y), fmaf(a.x, b.y, a.y * b.x));
}

// e^{SIGN * 2*pi*i * num / NFFT}, exact integer phase reduction.
template <int SIGN>
__device__ __forceinline__ float2 twd_outer(int num) {
  float ang = (float)SIGN * 6.28318530717958647692f *
              (float)(num & NFFT_MASK) * (1.0f / (float)NFFT);
  float s, c;
  sincosf(ang, &s, &c);
  return make_float2(c, s);
}

// ---------------------------------------------------------------------------
// 16 independent 1024-point FFTs resident in LDS (Stockham radix-4,
// auto-sorting, ping-pong buffers). Element e of this thread's line lives at
// buf[line_off + e*estride]. Each thread owns jbase (0..15) and loops the
// 256 butterflies-per-line in strides of 16. SIGN=-1 forward, +1 inverse
// (unnormalized). Returns the buffer holding the result (5 stages -> bufB).
// ---------------------------------------------------------------------------
template <int SIGN>
__device__ float2* fft1024_lds(float2* bufA, float2* bufB,
                               int line_off, int estride, int jbase) {
  float2* src = bufA;
  float2* dst = bufB;
  int Ns = 1;
  for (int stage = 0; stage < 5; ++stage) {
    __syncthreads();
    for (int ji = 0; ji < 16; ++ji) {
      const int j = jbase + (ji << 4);          // 0..255
      float2 v0 = src[line_off + j * estride];
      float2 v1 = src[line_off + (j + 256) * estride];
      float2 v2 = src[line_off + (j + 512) * estride];
      float2 v3 = src[line_off + (j + 768) * estride];
      const int jm = j & (Ns - 1);
      if (jm != 0) {
        float ang = (float)SIGN * 6.28318530717958647692f *
                    (float)jm / (float)(Ns << 2);
        float s1, c1;
        sincosf(ang, &s1, &c1);
        float2 w1 = make_float2(c1, s1);
        float2 w2 = cmul(w1, w1);
        float2 w3 = cmul(w2, w1);
        v1 = cmul(v1, w1);
        v2 = cmul(v2, w2);
        v3 = cmul(v3, w3);
      }
      float2 t0 = cadd(v0, v2), t1 = csub(v0, v2);
      float2 t2 = cadd(v1, v3), t3 = csub(v1, v3);
      // SIGN * i * t3
      float2 sit3 = (SIGN < 0) ? make_float2(t3.y, -t3.x)
                               : make_float2(-t3.y, t3.x);
      const int jd = ((j - jm) << 2) + jm;      // (j/Ns)*4Ns + j%Ns
      dst[line_off + jd * estride]            = cadd(t0, t2);
      dst[line_off + (jd + Ns) * estride]     = cadd(t1, sit3);
      dst[line_off + (jd + 2 * Ns) * estride] = csub(t0, t2);
      dst[line_off + (jd + 3 * Ns) * estride] = csub(t1, sit3);
    }
    Ns <<= 2;
    float2* t = src; src = dst; dst = t;
  }
  __syncthreads();
  return src;
}

// ---------------------------------------------------------------------------
// Pass F1: padded load of x (left pad K_LEN zeros) / ir, column FFTs over n1
// (stride N2), then outer twiddle W^{-k1*n2}. 16 columns per block, 256 thr.
// ---------------------------------------------------------------------------
__global__ __launch_bounds__(256) void k_fwd_col(const float* __restrict__ x,
                                                 const float* __restrict__ ir,
                                                 float2* __restrict__ ws) {
  extern __shared__ float2 lds[];
  float2* bufA = lds;
  float2* bufB = lds + 16384;
  const int tid   = threadIdx.x;
  const int sig   = blockIdx.x >> 6;   // 0..35
  const int tile  = blockIdx.x & 63;
  const int cbase = tile << 4;

  for (int i = 0; i < 64; ++i) {
    const int idx = tid + (i << 8);                    // 0..16383
    const int e = idx >> 4, c = idx & 15;
    const long n = (long)e * N2 + (long)(cbase + c);   // time index of xp
    float val = 0.0f;
    if (sig < NSIG_X) {
      const long m = n - (long)K_LEN;                  // left zero-pad by K
      if (m >= 0 && m < T_LEN) val = x[(size_t)sig * T_LEN + m];
    } else {
      if (n < K_LEN) val = ir[(size_t)(sig - NSIG_X) * K_LEN + n];
    }
    bufA[(e << 4) + c] = make_float2(val, 0.0f);
  }

  float2* res = fft1024_lds<-1>(bufA, bufB, tid & 15, 16, tid >> 4);

  float2* __restrict__ g = ws + (size_t)sig * NFFT;
  for (int i = 0; i < 64; ++i) {
    const int idx = tid + (i << 8);
    const int e = idx >> 4, c = idx & 15;
    const int n2 = cbase + c;
    float2 v = cmul(res[(e << 4) + c], twd_outer<-1>(e * n2));
    g[(size_t)e * N2 + n2] = v;
  }
}

// ---------------------------------------------------------------------------
// Pass F2 / I1: contiguous 1024-pt row (I)FFTs, 16 rows per block, in place.
// Tile staging global->LDS uses the ASYNCcnt DMA path when available.
// ---------------------------------------------------------------------------
template <int SIGN>
__global__ __launch_bounds__(256) void k_row_fft(float2* __restrict__ ws) {
  extern __shared__ float2 lds[];
  float2* bufA = lds;
  float2* bufB = lds + 16 * 1025;                 // pad row stride -> no bank conflicts
  const int tid  = threadIdx.x;
  const int sig  = blockIdx.x >> 6;
  const int tile = blockIdx.x & 63;
  float2* __restrict__ g = ws + (size_t)sig * NFFT + ((size_t)tile << 4) * N2;

  if (blockIdx.x + 1 < gridDim.x)
    __builtin_prefetch((const void*)(g + 16384), 0, 1);  // global_prefetch_b8

#if HAVE_ASYNC_LDS
  for (int i = 0; i < 64; ++i) {
    const int idx = tid + (i << 8);               // fully coalesced
    ASYNC_LOAD_B64(g + idx, &bufA[(idx >> 10) * 1025 + (idx & 1023)]);
  }
  __builtin_amdgcn_s_wait_asynccnt(0);
#else
  for (int i = 0; i < 64; ++i) {
    const int idx = tid + (i << 8);               // fully coalesced
    bufA[(idx >> 10) * 1025 + (idx & 1023)] = g[idx];
  }
#endif

  float2* res = fft1024_lds<SIGN>(bufA, bufB, (tid >> 4) * 1025, 1, tid & 15);
  for (int i = 0; i < 64; ++i) {
    const int idx = tid + (i << 8);
    g[idx] = res[(idx >> 10) * 1025 + (idx & 1023)];
  }
}

// ---------------------------------------------------------------------------
// Pointwise stage as WMMA F32 16x16x4 per frequency bin:
//   A (16x4):  row b = batch, K = [Xr(ci0), Xi(ci0), Xr(ci1), Xi(ci1)]
//   B (4x16):  col n: n=2o -> Yr[o], n=2o+1 -> Yi[o] (conj(W)); cols 4..15 zero
//   D (16x16): D[b][2o]=Yr, D[b][2o+1]=Yi.
// Fragment layouts per ISA 7.12.2: for both A (16x4) and B (4x16) f32, the
// low half of the K range sits in lanes 0-15 and the high half in lanes 16-31:
//   A: lane<16 row l holds (K0,K1); lane>=16 row l-16 holds (K2,K3).
//   B: lane<16 col n holds rows (K0,K1); lane>=16 col n-16 holds rows (K2,K3).
// D: VGPR r -> lanes0-15 M=r, lanes16-31 M=r+8. 64 bins per block staged
// through LDS so global traffic stays coalesced; in-place overwrite of X by Y.
// ---------------------------------------------------------------------------
__global__ __launch_bounds__(256) void k_einsum(float2* __restrict__ ws) {
  __shared__ float2 Xt[32][65];
  __shared__ float2 Wt[4][65];
  __shared__ float2 Yt[32][65];
  __shared__ float  Dst[8][8][32];   // per-wave D staging

  const int tid  = threadIdx.x;
  const int lane = tid & 31;
  const int wv   = tid >> 5;
  const long f0  = (long)blockIdx.x * 64;
  const float2* __restrict__ Wsig = ws + (size_t)NSIG_X * NFFT;

#if HAVE_ASYNC_LDS
  for (int i = 0; i < 8; ++i) {                 // X tile [32 sig][64 f]
    const int idx = tid + (i << 8);
    const int s = idx >> 6, fo = idx & 63;
    ASYNC_LOAD_B64(ws + (size_t)s * NFFT + f0 + fo, &Xt[s][fo]);
  }
  {                                             // W tile [4][64]
    const int s = tid >> 6, fo = tid & 63;
    ASYNC_LOAD_B64(Wsig + (size_t)s * NFFT + f0 + fo, &Wt[s][fo]);
  }
  __builtin_amdgcn_s_wait_asynccnt(0);
#else
  for (int i = 0; i < 8; ++i) {                 // X tile [32 sig][64 f]
    const int idx = tid + (i << 8);
    const int s = idx >> 6, fo = idx & 63;
    Xt[s][fo] = ws[(size_t)s * NFFT + f0 + fo];
  }
  {                                             // W tile [4][64]
    const int s = tid >> 6, fo = tid & 63;
    Wt[s][fo] = Wsig[(size_t)s * NFFT + f0 + fo];
  }
#endif
  __syncthreads();

  const int bb   = lane & 15;                         // unpack: batch
  const int oo   = lane >> 4;                         // unpack: out channel
  const int sA   = ((lane & 15) << 1) + (lane >> 4);  // A-frag signal b*2+ci
  const int nB   = lane & 15;                         // B-frag column
  const int half = lane >> 4;                         // K half == input channel

  for (int k = 0; k < 8; ++k) {
    const int fo = (wv << 3) + k;
    const float2 xv = Xt[sA][fo];
    v2f aF = {xv.x, xv.y};
    v2f bF = {0.0f, 0.0f};
    if (nB < 4) {
      const int o = nB >> 1, odd = nB & 1;
      // this lane half's two K rows are (2*half, 2*half+1) = (Wr,Wi) of ci=half
      const float2 wc = Wt[(o << 1) + half][fo];      // W[o][ci=half][f]
      bF[0] = odd ? -wc.y : wc.x;   // even K row: n even -> Wr, n odd -> -Wi
      bF[1] = odd ?  wc.x : wc.y;   // odd  K row: n even -> Wi, n odd ->  Wr
    }
    v8f cF = {};
    v8f d = __builtin_amdgcn_wmma_f32_16x16x4_f32(false, aF, false, bF,
                                                  (short)0, cF, false, false);
#pragma unroll
    for (int v = 0; v < 8; ++v) Dst[wv][v][lane] = d[v];
    // element (M,N) sits at Dst[wv][M&7][N + 16*(M>>3)] (wave-private, DS in-order)
    const float yr = Dst[wv][bb & 7][(oo << 1) + ((bb >> 3) << 4)];
    const float yi = Dst[wv][bb & 7][(oo << 1) + 1 + ((bb >> 3) << 4)];
    Yt[(bb << 1) + oo][fo] = make_float2(yr, yi);
  }
  __syncthreads();

#if HAVE_ASYNC_LDS
  for (int i = 0; i < 8; ++i) {
    const int idx = tid + (i << 8);
    const int s = idx >> 6, fo = idx & 63;
    ASYNC_STORE_B64(ws + (size_t)s * NFFT + f0 + fo, &Yt[s][fo]);
  }
  __builtin_amdgcn_s_wait_asynccnt(0);
#else
  for (int i = 0; i < 8; ++i) {
    const int idx = tid + (i << 8);
    const int s = idx >> 6, fo = idx & 63;
    ws[(size_t)s * NFFT + f0 + fo] = Yt[s][fo];
  }
#endif
}

// ---------------------------------------------------------------------------
// Pass I2: twiddle W^{+k1*n2}, column IFFT over k1, scale 1/NFFT, emit y[n].
// ---------------------------------------------------------------------------
__global__ __launch_bounds__(256) void k_inv_col(const float2* __restrict__ wsIn,
                                                 float* __restrict__ out) {
  extern __shared__ float2 lds[];
  float2* bufA = lds;
  float2* bufB = lds + 16384;
  const int tid   = threadIdx.x;
  const int sig   = blockIdx.x >> 6;   // 0..31 -> (b,o)
  const int tile  = blockIdx.x & 63;
  const int cbase = tile << 4;
  const float2* __restrict__ g = wsIn + (size_t)sig * NFFT;

  if (blockIdx.x + 1 < gridDim.x)
    __builtin_prefetch((const void*)(g + (size_t)16 * N2), 0, 1);

  for (int i = 0; i < 64; ++i) {
    const int idx = tid + (i << 8);
    const int e = idx >> 4, c = idx & 15;
    const int n2 = cbase + c;
    float2 v = g[(size_t)e * N2 + n2];
    bufA[(e << 4) + c] = cmul(v, twd_outer<1>(e * n2));
  }
  float2* res = fft1024_lds<1>(bufA, bufB, tid & 15, 16, tid >> 4);

  const float scale = 1.0f / (float)NFFT;
  float* __restrict__ o = out + (size_t)sig * OUT_LEN;
  for (int i = 0; i < 64; ++i) {
    const int idx = tid + (i << 8);
    const int e = idx >> 4, c = idx & 15;
    const long n = (long)e * N2 + (long)(cbase + c);
    if (n < OUT_LEN) o[n] = res[(e << 4) + c].x * scale;
  }
}

// ---------------------------------------------------------------------------
extern "C" void kernel_launch(void* const* d_in, const int* in_sizes, int n_in,
                              void* d_out, int out_size, void* d_ws, size_t ws_size,
                              hipStream_t stream) {
  (void)in_sizes; (void)n_in; (void)out_size; (void)ws_size;
  const float* x  = (const float*)d_in[0];   // [16,2,480000] f32
  const float* ir = (const float*)d_in[1];   // [2,2,143999]  f32 (pre-flipped)
  float*  out = (float*)d_out;               // [16,2,480001] f32
  float2* ws  = (float2*)d_ws;               // needs 36 * 2^20 * 8 B = 288 MiB

  const int ldsCol = 2 * 16384 * (int)sizeof(float2);     // 256 KiB
  const int ldsRow = 2 * 16 * 1025 * (int)sizeof(float2); // ~256 KiB

  // Allow >64KB dynamic LDS (CDNA5 WGP has 320KB). Ignore errors: no-op on AMD.
  (void)hipFuncSetAttribute((const void*)k_fwd_col,
                            hipFuncAttributeMaxDynamicSharedMemorySize, ldsCol);
  (void)hipFuncSetAttribute((const void*)(k_row_fft<-1>),
                            hipFuncAttributeMaxDynamicSharedMemorySize, ldsRow);
  (void)hipFuncSetAttribute((const void*)(k_row_fft<1>),
                            hipFuncAttributeMaxDynamicSharedMemorySize, ldsRow);
  (void)hipFuncSetAttribute((const void*)k_inv_col,
                            hipFuncAttributeMaxDynamicSharedMemorySize, ldsCol);

  k_fwd_col<<<dim3(NSIG * 64), dim3(256), ldsCol, stream>>>(x, ir, ws);
  k_row_fft<-1><<<dim3(NSIG * 64), dim3(256), ldsRow, stream>>>(ws);
  k_einsum<<<dim3(NFFT / 64), dim3(256), 0, stream>>>(ws);
  k_row_fft<1><<<dim3(NSIG_X * 64), dim3(256), ldsRow, stream>>>(ws);
  k_inv_col<<<dim3(NSIG_X * 64), dim3(256), ldsCol, stream>>>(ws, out);
}